// Model_90718299226352
// MI455X (gfx1250) — compile-verified
//
#include <hip/hip_runtime.h>
#include <math.h>

#define IMAGE_SIZE 256
#define EPS 1e-6f
#define SIGMA 0.01f
#define VIEW_ANGLE (30.0f * 3.14159265358979323846f / 180.0f)

typedef __attribute__((ext_vector_type(2))) float v2f;
typedef __attribute__((ext_vector_type(8))) float v8f;

// ---------------------------------------------------------------------------
// Setup: transform + project vertices, build per-edge coefficient table
// B[(edge*3 + k) * NPAD + face],  k = 0:nx', 1:ny', 2:-d'  (all * sign/sigma)
// Also zeroes the output accumulator.
// ---------------------------------------------------------------------------
__global__ void setup_kernel(const float* __restrict__ verts,
                             const int* __restrict__ faces,
                             const float* __restrict__ cam,
                             float* __restrict__ B,
                             float* __restrict__ out,
                             int V, int F, int NPAD) {
    __shared__ float projx[1024];
    __shared__ float projy[1024];

    const int t = threadIdx.x;
    if (t == 0) out[0] = 0.0f;

    // look-at rotation (redundant per-thread; trivial cost)
    float ex = cam[0], ey = cam[1], ez = cam[2];
    float zx = -ex, zy = -ey, zz = -ez;
    float zn = sqrtf(zx * zx + zy * zy + zz * zz) + EPS;
    zx /= zn; zy /= zn; zz /= zn;
    // x = cross(up=(0,1,0), z) = (zz, 0, -zx)
    float xx = zz, xy = 0.0f, xz = -zx;
    float xn = sqrtf(xx * xx + xy * xy + xz * xz) + EPS;
    xx /= xn; xy /= xn; xz /= xn;
    // y = cross(z, x)
    float yx = zy * xz - zz * xy;
    float yy = zz * xx - zx * xz;
    float yz = zx * xy - zy * xx;
    const float w = tanf(VIEW_ANGLE);

    for (int i = t; i < V; i += blockDim.x) {
        float px = verts[3 * i + 0] - ex;
        float py = verts[3 * i + 1] - ey;
        float pz = verts[3 * i + 2] - ez;
        float vx = xx * px + xy * py + xz * pz;
        float vy = yx * px + yy * py + yz * pz;
        float vz = zx * px + zy * py + zz * pz;
        float inv = 1.0f / (vz * w);
        projx[i] = vx * inv;
        projy[i] = vy * inv;
    }
    __syncthreads();

    for (int f = t; f < NPAD; f += blockDim.x) {
        if (f < F) {
            int i0 = faces[3 * f + 0], i1 = faces[3 * f + 1], i2 = faces[3 * f + 2];
            float ax[3] = {projx[i0], projx[i1], projx[i2]};
            float ay[3] = {projy[i0], projy[i1], projy[i2]};
            float e01x = ax[1] - ax[0], e01y = ay[1] - ay[0];
            float e02x = ax[2] - ax[0], e02y = ay[2] - ay[0];
            float area = e01x * e02y - e01y * e02x;
            float s = (area > 0.0f) ? 1.0f : ((area < 0.0f) ? -1.0f : 0.0f);
            float sc = s * (1.0f / SIGMA);
            const int ea[3] = {0, 1, 2};
            const int eb[3] = {1, 2, 0};
#pragma unroll
            for (int e = 0; e < 3; ++e) {
                float Ax = ax[ea[e]], Ay = ay[ea[e]];
                float Bx = ax[eb[e]], By = ay[eb[e]];
                float Ex = Bx - Ax, Ey = By - Ay;
                float en = sqrtf(Ex * Ex + Ey * Ey) + EPS;
                float nx = -Ey / en, ny = Ex / en;
                float d = Ax * nx + Ay * ny;
                B[(e * 3 + 0) * NPAD + f] = nx * sc;
                B[(e * 3 + 1) * NPAD + f] = ny * sc;
                B[(e * 3 + 2) * NPAD + f] = -d * sc;
            }
        } else {
#pragma unroll
            for (int e = 0; e < 3; ++e) {
                B[(e * 3 + 0) * NPAD + f] = 0.0f;
                B[(e * 3 + 1) * NPAD + f] = 0.0f;
                B[(e * 3 + 2) * NPAD + f] = 0.0f;
            }
        }
    }
}

__device__ __forceinline__ float sigm(float t) {
    // 1 / (1 + e^-t): v_exp_f32 + v_rcp_f32, overflow saturates correctly (->0)
    return __builtin_amdgcn_rcpf(1.0f + __expf(-t));
}

// ---------------------------------------------------------------------------
// Main: one wave32 per 16-pixel tile; per face-tile of 16 faces issue
// 3x V_WMMA_F32_16X16X4_F32 (one per triangle edge) producing the 16x16
// sigmoid-argument blocks; VALU/TRANS does sigmoids + product fold, which
// co-executes with the matrix pipe.
// ---------------------------------------------------------------------------
__global__ void sil_loss_kernel(const float* __restrict__ B,
                                const float* __restrict__ ref,
                                float* __restrict__ out,
                                int F, int NPAD) {
    const int lane = threadIdx.x & 31;
    const int wave = (blockIdx.x * blockDim.x + threadIdx.x) >> 5;
    const int half = lane >> 4;   // 0: lanes 0-15, 1: lanes 16-31
    const int lid  = lane & 15;
    const int pixBase = wave * 16;

    // A matrix 16x4 f32: lane<16 holds (K=0,K=1)=(x,y); lane>=16 holds (K=2,K=3)=(1,0)
    v2f a;
    if (half == 0) {
        int p  = pixBase + lid;
        int xi = p & (IMAGE_SIZE - 1);
        int yi = p >> 8;
        a.x = (2.0f * (float)xi + 1.0f) / (float)IMAGE_SIZE - 1.0f;
        a.y = 1.0f - (2.0f * (float)yi + 1.0f) / (float)IMAGE_SIZE;
    } else {
        a.x = 1.0f;
        a.y = 0.0f;
    }

    float prod[8];
#pragma unroll
    for (int r = 0; r < 8; ++r) prod[r] = 1.0f;

    const int ntiles = NPAD >> 4;
    for (int t = 0; t < ntiles; ++t) {
        const int n = t * 16 + lid;
        // B matrix 4x16 f32 per edge: lane<16 holds (K=0,K=1); lane>=16 holds (K=2,K=3=0)
        v2f b0, b1, b2;
        if (half == 0) {
            b0.x = B[(0 * 3 + 0) * NPAD + n]; b0.y = B[(0 * 3 + 1) * NPAD + n];
            b1.x = B[(1 * 3 + 0) * NPAD + n]; b1.y = B[(1 * 3 + 1) * NPAD + n];
            b2.x = B[(2 * 3 + 0) * NPAD + n]; b2.y = B[(2 * 3 + 1) * NPAD + n];
        } else {
            b0.x = B[(0 * 3 + 2) * NPAD + n]; b0.y = 0.0f;
            b1.x = B[(1 * 3 + 2) * NPAD + n]; b1.y = 0.0f;
            b2.x = B[(2 * 3 + 2) * NPAD + n]; b2.y = 0.0f;
        }

        v8f c = {};
        v8f d0 = __builtin_amdgcn_wmma_f32_16x16x4_f32(false, a, false, b0, (short)0, c, false, false);
        v8f d1 = __builtin_amdgcn_wmma_f32_16x16x4_f32(false, a, false, b1, (short)0, c, false, false);
        v8f d2 = __builtin_amdgcn_wmma_f32_16x16x4_f32(false, a, false, b2, (short)0, c, false, false);

        const bool valid = n < F;   // mask zero-padded faces
#pragma unroll
        for (int r = 0; r < 8; ++r) {
            float s0 = sigm(d0[r]);
            float s1 = sigm(d1[r]);
            float s2 = sigm(d2[r]);
            float frag = valid ? (s0 * s1 * s2) : 0.0f;
            prod[r] *= (1.0f - frag);
        }
    }

    // Product reduction over the 16 faces/lanes within each half-wave.
    // D layout: lanes 0-15 hold rows M=0..7 (VGPR r), lanes 16-31 hold M=8..15.
#pragma unroll
    for (int off = 1; off < 16; off <<= 1) {
#pragma unroll
        for (int r = 0; r < 8; ++r) prod[r] *= __shfl_xor(prod[r], off, 32);
    }

    if (lid == 0) {  // one writer per half-wave (pixels pixBase+half*8 .. +7)
        float acc = 0.0f;
#pragma unroll
        for (int r = 0; r < 8; ++r) {
            int m = pixBase + half * 8 + r;
            float sil  = 1.0f - prod[r];
            float diff = sil - ref[m];
            acc += diff * diff;
        }
        atomicAdd(out, acc);
    }
}

// ---------------------------------------------------------------------------
extern "C" void kernel_launch(void* const* d_in, const int* in_sizes, int n_in,
                              void* d_out, int out_size, void* d_ws, size_t ws_size,
                              hipStream_t stream) {
    const float* verts = (const float*)d_in[0];   // (1, V, 3) f32
    const int*   faces = (const int*)d_in[1];     // (1, F, 3) i32
    const float* cam   = (const float*)d_in[2];   // (3,) f32
    const float* ref   = (const float*)d_in[3];   // (1, 256, 256) f32
    float* out = (float*)d_out;                   // scalar loss
    float* B   = (float*)d_ws;                    // 9 * NPAD floats (~36 KB)

    const int V = in_sizes[0] / 3;
    const int F = in_sizes[1] / 3;
    const int NPAD = (F + 15) & ~15;

    setup_kernel<<<1, 1024, 0, stream>>>(verts, faces, cam, B, out, V, F, NPAD);

    const int numPix   = IMAGE_SIZE * IMAGE_SIZE;     // 65536
    const int numWaves = numPix / 16;                 // 4096 waves
    const int block    = 256;                         // 8 waves/block (wave32)
    const int grid     = (numWaves * 32) / block;     // 512 blocks
    sil_loss_kernel<<<grid, block, 0, stream>>>(B, ref, out, F, NPAD);
}